// EdgeConv_76596446757083
// MI455X (gfx1250) — compile-verified
//
#include <hip/hip_runtime.h>
#include <math.h>

typedef float v2f __attribute__((ext_vector_type(2)));
typedef float v8f __attribute__((ext_vector_type(8)));

#define B_ 8
#define C_ 64
#define N_ 4096
#define O_ 64
#define K_ 20
#define TWOC 128
#define EPS_ 1e-5f
#define NEG_SLOPE_ 0.2f
#define CHUNK 256
#define DSTRIDE 260   // 16-row LDS dist buffer stride (bank-conflict padding)

// ---------------- transpose x[B,C,N] -> XT[B,N,C] ----------------
__global__ __launch_bounds__(256)
void transpose_kernel(const float* __restrict__ x, float* __restrict__ xt) {
  __shared__ float tile[32][33];
  int b = blockIdx.z;
  int nBase = blockIdx.x * 32;
  int cBase = blockIdx.y * 32;
  int tx = threadIdx.x, ty = threadIdx.y;   // 32 x 8
#pragma unroll
  for (int i = 0; i < 4; ++i) {
    int c = cBase + ty + 8 * i;
    tile[ty + 8 * i][tx] = x[((size_t)(b * C_ + c)) * N_ + nBase + tx];
  }
  __syncthreads();
#pragma unroll
  for (int i = 0; i < 4; ++i) {
    int n = nBase + ty + 8 * i;
    xt[((size_t)(b * N_ + n)) * C_ + cBase + tx] = tile[tx][ty + 8 * i];
  }
}

// ---------------- xx[b,n] = sum_c XT[b,n,c]^2 ----------------
__global__ __launch_bounds__(256)
void xx_kernel(const float* __restrict__ xt, float* __restrict__ xx) {
  int i = blockIdx.x * 256 + threadIdx.x;   // over B*N
  const float4* row = (const float4*)(xt + (size_t)i * C_);
  float s = 0.f;
#pragma unroll
  for (int j = 0; j < 16; ++j) {
    float4 v = row[j];
    s += v.x * v.x + v.y * v.y + v.z * v.z + v.w * v.w;
  }
  xx[i] = s;
}

// ---------------- Z = XT*W1^T, T = XT*(W2-W1)^T via f32 WMMA ----------------
// A 16x4 f32 frag: lane l holds A[m=l%16][k=2*(l/16)+j] in component j.
// B 4x16 f32 frag: lane l holds B[k=2*(l/16)+j][n=l%16] (symmetric layout).
// D 16x16 f32: VGPR r, lanes 0-15 -> M=r, lanes 16-31 -> M=r+8; N=l%16.
__global__ __launch_bounds__(256)
void zt_kernel(const float* __restrict__ xt, const float* __restrict__ W,
               float* __restrict__ Z, float* __restrict__ T) {
  int lane = threadIdx.x & 31;
  int wave = threadIdx.x >> 5;
  int m = lane & 15, hi = lane >> 4;
  int b = blockIdx.x >> 5;                      // 32 blocks per batch
  int nBase = (blockIdx.x & 31) * 128 + wave * 16;

  const float* arow = xt + ((size_t)(b * N_ + nBase + m)) * C_ + 2 * hi;
  v2f a[16];
#pragma unroll
  for (int i = 0; i < 16; ++i) a[i] = *(const v2f*)(arow + 4 * i);

#pragma unroll
  for (int ot = 0; ot < 4; ++ot) {
    int obase = ot * 16;
    const float* wrow = W + (size_t)(obase + m) * TWOC + 2 * hi;
    v8f accZ = {0, 0, 0, 0, 0, 0, 0, 0};
    v8f accT = {0, 0, 0, 0, 0, 0, 0, 0};
#pragma unroll
    for (int i = 0; i < 16; ++i) {
      v2f b1 = *(const v2f*)(wrow + 4 * i);        // W1 = W[:, 0:64]
      v2f b2 = *(const v2f*)(wrow + 64 + 4 * i);   // W2 = W[:, 64:128]
      v2f bd = b2 - b1;
      accZ = __builtin_amdgcn_wmma_f32_16x16x4_f32(false, a[i], false, b1,
                                                   (short)0, accZ, false, false);
      accT = __builtin_amdgcn_wmma_f32_16x16x4_f32(false, a[i], false, bd,
                                                   (short)0, accT, false, false);
    }
#pragma unroll
    for (int r = 0; r < 8; ++r) {
      int row = nBase + r + 8 * hi;
      size_t off = ((size_t)(b * N_ + row)) * O_ + obase + m;
      Z[off] = accZ[r];
      T[off] = accT[r];
    }
  }
}

// ---------------- fused pairwise-distance + top-K ----------------
// Block: 16 query rows of one batch, 256 threads (8 waves).
// Ranking key: 2*inner[i][j] - xx[j]  (per-row constant -xx[i] dropped).
__global__ __launch_bounds__(256)
void knn_kernel(const float* __restrict__ xt, const float* __restrict__ xx,
                int* __restrict__ idxOut) {
  __shared__ float sd[16 * DSTRIDE];
  __shared__ float cv[16 * 16 * K_];
  __shared__ int   ci[16 * 16 * K_];

  int t = threadIdx.x;
  int lane = t & 31, wave = t >> 5;
  int m = lane & 15, hi = lane >> 4;
  int b = blockIdx.x >> 8;                  // 256 tiles per batch
  int qBase = (blockIdx.x & 255) * 16;

  const float* arow = xt + ((size_t)(b * N_ + qBase + m)) * C_ + 2 * hi;
  v2f a[16];
#pragma unroll
  for (int i = 0; i < 16; ++i) a[i] = *(const v2f*)(arow + 4 * i);

  float bestV[K_];
  int bestI[K_];
#pragma unroll
  for (int j = 0; j < K_; ++j) { bestV[j] = -3.0e38f; bestI[j] = 0; }

  int row = t & 15;     // selection: row owned
  int slice = t >> 4;   // selection: 16-column slice owned

  for (int cc = 0; cc < N_ / CHUNK; ++cc) {
    int chunkBase = cc * CHUNK;
    // ---- 8 waves compute 16 x 256 ranking-key chunk via WMMA ----
#pragma unroll
    for (int tt2 = 0; tt2 < 2; ++tt2) {
      int tile = wave * 2 + tt2;
      int colBase = chunkBase + tile * 16;
      const float* brow = xt + ((size_t)(b * N_ + colBase + m)) * C_ + 2 * hi;
      v8f acc = {0, 0, 0, 0, 0, 0, 0, 0};
#pragma unroll
      for (int i = 0; i < 16; ++i) {
        v2f bf = *(const v2f*)(brow + 4 * i);
        acc = __builtin_amdgcn_wmma_f32_16x16x4_f32(false, a[i], false, bf,
                                                    (short)0, acc, false, false);
      }
      float xxj = xx[b * N_ + colBase + m];
#pragma unroll
      for (int r = 0; r < 8; ++r) {
        int rr = r + 8 * hi;
        sd[rr * DSTRIDE + tile * 16 + m] = 2.f * acc[r] - xxj;
      }
    }
    __syncthreads();
    // ---- each thread scans its 16 columns, keeps sorted top-K in regs ----
#pragma unroll
    for (int c = 0; c < 16; ++c) {
      float d = sd[row * DSTRIDE + slice * 16 + c];
      int j = chunkBase + slice * 16 + c;
      if (d > bestV[K_ - 1]) {
        float v = d; int vi = j;
#pragma unroll
        for (int p = 0; p < K_; ++p) {       // bubble-insert, keeps list sorted desc
          bool gt = v > bestV[p];
          float tv = bestV[p]; int ti = bestI[p];
          bestV[p] = gt ? v : tv; bestI[p] = gt ? vi : ti;
          v = gt ? tv : v; vi = gt ? ti : vi;
        }
      }
    }
    __syncthreads();
  }

  // ---- dump 16 sorted lists per row, merge with head pointers ----
#pragma unroll
  for (int p = 0; p < K_; ++p) {
    cv[(row * 16 + slice) * K_ + p] = bestV[p];
    ci[(row * 16 + slice) * K_ + p] = bestI[p];
  }
  __syncthreads();

  if (t < 16) {
    int r2 = t;
    int heads[16];
#pragma unroll
    for (int s = 0; s < 16; ++s) heads[s] = 0;
    for (int kk = 0; kk < K_; ++kk) {
      float bv = -3.4e38f; int bs = 0; int bi = 0;
#pragma unroll
      for (int s = 0; s < 16; ++s) {
        int h = heads[s];
        if (h < K_) {
          float v = cv[(r2 * 16 + s) * K_ + h];
          if (v > bv) { bv = v; bs = s; bi = ci[(r2 * 16 + s) * K_ + h]; }
        }
      }
#pragma unroll
      for (int s = 0; s < 16; ++s) heads[s] += (s == bs) ? 1 : 0;
      idxOut[((size_t)(b * N_ + qBase + r2)) * K_ + kk] = bi;
    }
  }
}

// ---------------- per-channel sum / sumsq of y = Z[idx] + T ----------------
__global__ __launch_bounds__(64)
void stats_kernel(const float* __restrict__ Z, const float* __restrict__ T,
                  const int* __restrict__ idx, float* __restrict__ S) {
  int o = threadIdx.x;
  int b = blockIdx.x >> 5;
  int nBase = (blockIdx.x & 31) * 128;
  float fs = 0.f, fq = 0.f;
  for (int nn = 0; nn < 128; ++nn) {
    int n = nBase + nn;
    float to = T[((size_t)(b * N_ + n)) * O_ + o];
    const int* ip = idx + ((size_t)(b * N_ + n)) * K_;
#pragma unroll
    for (int k = 0; k < K_; ++k) {
      int id = ip[k];
      float y = Z[((size_t)(b * N_ + id)) * O_ + o] + to;
      fs += y; fq += y * y;
    }
  }
  atomicAdd(&S[o], fs);
  atomicAdd(&S[64 + o], fq);
}

__global__ void init_kernel(float* S) { S[threadIdx.x] = 0.f; }

__global__ void finalize_kernel(const float* __restrict__ S,
                                const float* __restrict__ gamma,
                                const float* __restrict__ beta,
                                float* __restrict__ SS) {
  int o = threadIdx.x;  // 64
  const float cnt = (float)B_ * (float)N_ * (float)K_;
  float mean = S[o] / cnt;
  float var = S[64 + o] / cnt - mean * mean;
  float scale = gamma[o] * rsqrtf(var + EPS_);
  SS[o] = scale;
  SS[64 + o] = beta[o] - mean * scale;
}

// ---------------- apply BN + LeakyReLU + max over K, write [B,O,N] ----------------
__global__ __launch_bounds__(64)
void out_kernel(const float* __restrict__ Z, const float* __restrict__ T,
                const int* __restrict__ idx, const float* __restrict__ SS,
                float* __restrict__ out) {
  int o = threadIdx.x;
  int b = blockIdx.x >> 5;
  int nBase = (blockIdx.x & 31) * 128;
  float scale = SS[o], shift = SS[64 + o];
  for (int nn = 0; nn < 128; ++nn) {
    int n = nBase + nn;
    float to = T[((size_t)(b * N_ + n)) * O_ + o];
    const int* ip = idx + ((size_t)(b * N_ + n)) * K_;
    float mx = -3.4e38f;
#pragma unroll
    for (int k = 0; k < K_; ++k) {
      int id = ip[k];
      float y = Z[((size_t)(b * N_ + id)) * O_ + o] + to;
      float v = y * scale + shift;
      v = (v >= 0.f) ? v : NEG_SLOPE_ * v;
      mx = fmaxf(mx, v);
    }
    out[((size_t)(b * O_ + o)) * N_ + n] = mx;
  }
}

extern "C" void kernel_launch(void* const* d_in, const int* in_sizes, int n_in,
                              void* d_out, int out_size, void* d_ws, size_t ws_size,
                              hipStream_t stream) {
  const float* x     = (const float*)d_in[0];   // [B, C, N]
  const float* W     = (const float*)d_in[1];   // [O, 2C]
  const float* gamma = (const float*)d_in[2];   // [O]
  const float* beta  = (const float*)d_in[3];   // [O]
  float* out = (float*)d_out;                   // [B, O, N]

  float* ws = (float*)d_ws;
  float* XT  = ws;                                  // B*N*C
  float* XX  = XT + (size_t)B_ * N_ * C_;           // B*N
  float* Z   = XX + (size_t)B_ * N_;                // B*N*O
  float* T   = Z + (size_t)B_ * N_ * O_;            // B*N*O
  int*   IDX = (int*)(T + (size_t)B_ * N_ * O_);    // B*N*K
  float* S   = (float*)(IDX + (size_t)B_ * N_ * K_);// 128
  float* SS  = S + 128;                             // 128

  transpose_kernel<<<dim3(N_ / 32, C_ / 32, B_), dim3(32, 8), 0, stream>>>(x, XT);
  xx_kernel<<<(B_ * N_) / 256, 256, 0, stream>>>(XT, XX);
  zt_kernel<<<B_ * (N_ / 128), 256, 0, stream>>>(XT, W, Z, T);
  knn_kernel<<<B_ * (N_ / 16), 256, 0, stream>>>(XT, XX, IDX);
  init_kernel<<<1, 128, 0, stream>>>(S);
  stats_kernel<<<B_ * (N_ / 128), 64, 0, stream>>>(Z, T, IDX, S);
  finalize_kernel<<<1, 64, 0, stream>>>(S, gamma, beta, SS);
  out_kernel<<<B_ * (N_ / 128), 64, 0, stream>>>(Z, T, IDX, SS, out);
}